// qnn_graph_89489938579835
// MI455X (gfx1250) — compile-verified
//
#include <hip/hip_runtime.h>
#include <hip/hip_bf16.h>

// RWKV chunked wkv recurrence on MI455X (gfx1250), reformulated from a serial
// T=32 scan into dense 16x16 GEMM tiles executed with V_WMMA_F32_16X16X4_F32.
//
//  Per head h:
//    a_t[i]   = prod_{u<=t} td_u[i]            (tiny per-channel scan, 64 lanes)
//    Vt[t,i]  = v_t[i] / a_t[i]
//    S        = R K^T                 [32,32]  (WMMA)
//    out[t,i] = tf[i]*v_t[i]*S[t,t]
//             + a_{t-1}[i] * ( (tril_strict(S) @ Vt)[t,i] + (R @ state0^T)[t,i] )   (WMMA)
//    s_fin    = diag(a_31) * ( state0 + Vt^T @ K )                                  (WMMA)
//
// One 128-thread block (4 wave32) per head; all per-head operands staged in
// ~53KB of LDS; each wave computes independent 16x16 tiles (384 WMMA/head).

static constexpr int kT = 32;   // chunk length
static constexpr int kH = 4096; // heads
static constexpr int kD = 64;   // head size

typedef __attribute__((ext_vector_type(2))) float v2f;
typedef __attribute__((ext_vector_type(8))) float v8f;

__device__ __forceinline__ v8f wmma4(v2f a, v2f b, v8f c) {
    // D = A(16x4,f32) * B(4x16,f32) + C(16x16,f32)
    // args: (neg_a, A, neg_b, B, c_mod, C, reuse_a, reuse_b)
    return __builtin_amdgcn_wmma_f32_16x16x4_f32(false, a, false, b, (short)0, c,
                                                 false, false);
}

__global__ __launch_bounds__(128) void rwkv_wkv_chunk(
    const float* __restrict__ gk, const float* __restrict__ gv,
    const float* __restrict__ gr, const float* __restrict__ gstate,
    const float* __restrict__ gtd, const float* __restrict__ gtf,
    float* __restrict__ gout, float* __restrict__ gfstate)
{
    const int h    = blockIdx.x;
    const int tid  = threadIdx.x;
    const int lane = tid & 31;
    const int wv   = tid >> 5;

    __shared__ float s_k [kT * kD];   // K[t][j]
    __shared__ float s_r [kT * kD];   // R[t][j]
    __shared__ float s_v [kT * kD];   // V[t][i]  -> becomes Vt = V / a
    __shared__ float s_td[kT * kD];   // td[t][i] -> becomes a_t (cumprod)
    __shared__ float s_st[kD * kD];   // state0[i][j]
    __shared__ float s_s [kT * kT];   // S[t][tau]
    __shared__ float s_tf[kD];

    // ---- stage per-head operands into LDS (coalesced: rows are contiguous) ----
    for (int idx = tid; idx < kT * kD; idx += 128) {
        const int t = idx >> 6, j = idx & 63;
        const size_t g = (size_t)(t * kH + h) * kD + j;
        s_k [idx] = gk [g];
        s_r [idx] = gr [g];
        s_v [idx] = gv [g];
        s_td[idx] = gtd[g];
    }
    for (int idx = tid; idx < kD * kD; idx += 128)
        s_st[idx] = gstate[(size_t)h * kD * kD + idx];
    if (tid < kD) s_tf[tid] = gtf[(size_t)h * kD + tid];
    __syncthreads();

    // ---- cumulative decay in place: s_td[t][i] := prod_{u<=t} td_u[i] ----
    if (tid < kD) {
        float run = 1.0f;
        for (int t = 0; t < kT; ++t) {
            run *= s_td[t * kD + tid];
            s_td[t * kD + tid] = run;
        }
    }
    __syncthreads();

    // ---- Vt = V / a ----
    for (int idx = tid; idx < kT * kD; idx += 128)
        s_v[idx] = s_v[idx] / s_td[idx];
    __syncthreads();

    // Fragment lane mapping (wave32 f32 WMMA):
    //   A 16x4 : M = lane&15, K = k0 + 2*(lane>>4) + {0,1}  (a.x, a.y)
    //   B 4x16 : N = lane&15, K = k0 + 2*(lane>>4) + {0,1}  (b.x, b.y)
    //   C 16x16: row = e + 8*(lane>>4), col = lane&15        (e = 0..7)
    const int fm    = lane & 15;          // M / N index within tile
    const int fkoff = (lane >> 4) << 1;   // K sub-offset
    const int rbase = (lane >> 4) * 8;    // C row base
    const int col   = lane & 15;          // C column

    // ---- Phase 1: S = R K^T   (4 tiles of 16x16, one per wave) ----
    {
        const int m0 = (wv >> 1) * 16, n0 = (wv & 1) * 16;
        v8f c = {};
        #pragma unroll
        for (int k0 = 0; k0 < kD; k0 += 4) {
            const int kk = k0 + fkoff;
            v2f a, b;
            a.x = s_r[(m0 + fm) * kD + kk];
            a.y = s_r[(m0 + fm) * kD + kk + 1];
            // B = K^T: B[kk][n] = K[n][kk]
            b.x = s_k[(n0 + fm) * kD + kk];
            b.y = s_k[(n0 + fm) * kD + kk + 1];
            c = wmma4(a, b, c);
        }
        #pragma unroll
        for (int e = 0; e < 8; ++e)
            s_s[(m0 + rbase + e) * kT + n0 + col] = c[e];
    }
    __syncthreads();

    // ---- Phase 2: out = tf*v*diag(S) + a_{t-1} * (tril_strict(S)@Vt + R@state0^T)
    // 8 tiles (2 x 4), each wave does 2.
    for (int tt = wv; tt < 8; tt += 4) {
        const int m0 = (tt >> 2) * 16, n0 = (tt & 3) * 16;
        v8f c = {};
        // (tril_strict(S)) @ Vt : K-reduction over tau = 0..31
        #pragma unroll
        for (int k0 = 0; k0 < kT; k0 += 4) {
            const int kk = k0 + fkoff;
            const int m  = m0 + fm;
            v2f a, b;
            a.x = (kk     < m) ? s_s[m * kT + kk    ] : 0.0f;  // strict lower mask
            a.y = (kk + 1 < m) ? s_s[m * kT + kk + 1] : 0.0f;
            b.x = s_v[(kk    ) * kD + n0 + fm];
            b.y = s_v[(kk + 1) * kD + n0 + fm];
            c = wmma4(a, b, c);
        }
        // + R @ state0^T : K-reduction over j = 0..63
        #pragma unroll
        for (int k0 = 0; k0 < kD; k0 += 4) {
            const int kk = k0 + fkoff;
            v2f a, b;
            a.x = s_r[(m0 + fm) * kD + kk];
            a.y = s_r[(m0 + fm) * kD + kk + 1];
            // B = state0^T: B[kk][n] = state0[n][kk]
            b.x = s_st[(n0 + fm) * kD + kk];
            b.y = s_st[(n0 + fm) * kD + kk + 1];
            c = wmma4(a, b, c);
        }
        // per-element fixup + store
        #pragma unroll
        for (int e = 0; e < 8; ++e) {
            const int t = m0 + rbase + e, i = n0 + col;
            const float a_prev = (t == 0) ? 1.0f : s_td[(t - 1) * kD + i];
            const float v_ti   = s_v[t * kD + i] * s_td[t * kD + i]; // reconstruct v
            const float val    = s_tf[i] * v_ti * s_s[t * kT + t] + a_prev * c[e];
            gout[(size_t)(t * kH + h) * kD + i] = val;
        }
    }

    // ---- Phase 3: s_final = diag(a_31) * (state0 + Vt^T @ K)
    // 16 tiles (4 x 4), each wave does 4. Reads only LDS written before Phase 1.
    for (int tt = wv; tt < 16; tt += 4) {
        const int m0 = (tt >> 2) * 16, n0 = (tt & 3) * 16;
        v8f c;
        #pragma unroll
        for (int e = 0; e < 8; ++e)               // C init = state0 tile
            c[e] = s_st[(m0 + rbase + e) * kD + n0 + col];
        #pragma unroll
        for (int k0 = 0; k0 < kT; k0 += 4) {
            const int kk = k0 + fkoff;
            v2f a, b;
            // A = Vt^T: A[m][kk] = Vt[kk][m]
            a.x = s_v[(kk    ) * kD + m0 + fm];
            a.y = s_v[(kk + 1) * kD + m0 + fm];
            b.x = s_k[(kk    ) * kD + n0 + fm];
            b.y = s_k[(kk + 1) * kD + n0 + fm];
            c = wmma4(a, b, c);
        }
        #pragma unroll
        for (int e = 0; e < 8; ++e) {
            const int i = m0 + rbase + e, j = n0 + col;
            gfstate[((size_t)h * kD + i) * kD + j] = s_td[(kT - 1) * kD + i] * c[e];
        }
    }
}

extern "C" void kernel_launch(void* const* d_in, const int* in_sizes, int n_in,
                              void* d_out, int out_size, void* d_ws, size_t ws_size,
                              hipStream_t stream) {
    const float* k  = (const float*)d_in[0];
    const float* v  = (const float*)d_in[1];
    const float* r  = (const float*)d_in[2];
    const float* st = (const float*)d_in[3];
    const float* td = (const float*)d_in[4];
    const float* tf = (const float*)d_in[5];

    float* out = (float*)d_out;                     // [T,H,D] first
    float* fs  = out + (size_t)kT * kH * kD;        // then [H,D,D]

    rwkv_wkv_chunk<<<kH, 128, 0, stream>>>(k, v, r, st, td, tf, out, fs);
}